// ReflexiveAttention_49649821941958
// MI455X (gfx1250) — compile-verified
//
#include <hip/hip_runtime.h>
#include <hip/hip_bf16.h>

namespace {

constexpr int kDim   = 512;
constexpr int kHeads = 8;
constexpr int kB     = 2;
constexpr int kL     = 1536;
constexpr int kFused = 24;           // kHeads * 3
constexpr int kPos   = kB * kL;      // 3072
constexpr int kBH    = kB * kHeads;  // 16
constexpr float kEps = 1e-8f;

typedef float v2f __attribute__((ext_vector_type(2)));
typedef float v8f __attribute__((ext_vector_type(8)));

// ---- workspace layout (in floats) ----
constexpr size_t OFF_R  = 0;                          // kPos * 9
constexpr size_t OFF_T  = OFF_R  + (size_t)kPos * 9;  // kPos * 3
constexpr size_t OFF_QR = OFF_T  + (size_t)kPos * 3;  // kBH*kL*4 (k padded to 4)
constexpr size_t OFF_KR = OFF_QR + (size_t)kBH * kL * 4;
constexpr size_t OFF_QD = OFF_KR + (size_t)kBH * kL * 4;
constexpr size_t OFF_KD = OFF_QD + (size_t)kBH * kL * 4;
constexpr size_t OFF_V  = OFF_KD + (size_t)kBH * kL * 4;
constexpr size_t OFF_QQ = OFF_V  + (size_t)kBH * kL * 4;  // kBH*kL
constexpr size_t OFF_KK = OFF_QQ + (size_t)kBH * kL;      // kBH*kL (unused by attn now)
constexpr size_t OFF_O  = OFF_KK + (size_t)kBH * kL;      // kPos*kFused

__device__ __forceinline__ void norm3(const float* v, float* o) {
  float nn  = sqrtf(v[0]*v[0] + v[1]*v[1] + v[2]*v[2]);
  float inv = 1.0f / fmaxf(nn, kEps);
  o[0] = v[0]*inv; o[1] = v[1]*inv; o[2] = v[2]*inv;
}

// Async global -> LDS copy, 16 bytes, per-lane addresses. Tracked by ASYNCcnt.
__device__ __forceinline__ void async_copy_b128(unsigned int lds_off, const void* gptr) {
  asm volatile("global_load_async_to_lds_b128 %0, %1, off"
               :: "v"(lds_off), "v"((unsigned long long)(size_t)gptr)
               : "memory");
}
__device__ __forceinline__ void wait_async_le2() {
  asm volatile("s_wait_asynccnt 0x2" ::: "memory");
}
__device__ __forceinline__ void wait_async_0() {
  asm volatile("s_wait_asynccnt 0x0" ::: "memory");
}
// Low 32 bits of a generic LDS pointer == wave LDS byte offset (flat->LDS truncates).
__device__ __forceinline__ unsigned int lds_offset(const void* p) {
  return (unsigned int)(size_t)p;
}

} // namespace

// ------------------------------------------------------------------
// Kernel 1: per-residue frames (Gram-Schmidt) -> R (3x3 rows), t
// ------------------------------------------------------------------
__global__ void ra_frames_kernel(const float* __restrict__ coords,
                                 const unsigned char* __restrict__ valid,
                                 float* __restrict__ ws) {
  int p = blockIdx.x * blockDim.x + threadIdx.x;
  if (p >= kPos) return;
  const float* cp = coords + (size_t)p * 12;

  float ca[3], vn[3], vc[3], vb[3];
#pragma unroll
  for (int d = 0; d < 3; ++d) {
    ca[d] = cp[3 + d];
    vn[d] = cp[d]     - ca[d];
    vc[d] = cp[6 + d] - ca[d];
    vb[d] = cp[9 + d] - ca[d];
  }
  float xb[3]; norm3(vn, xb);
  float dcx = vc[0]*xb[0] + vc[1]*xb[1] + vc[2]*xb[2];
  float y[3] = { vc[0] - dcx*xb[0], vc[1] - dcx*xb[1], vc[2] - dcx*xb[2] };
  float yb[3]; norm3(y, yb);
  float cr[3] = { xb[1]*yb[2] - xb[2]*yb[1],
                  xb[2]*yb[0] - xb[0]*yb[2],
                  xb[0]*yb[1] - xb[1]*yb[0] };
  float wb[3]; norm3(cr, wb);
  float dbw = vb[0]*wb[0] + vb[1]*wb[1] + vb[2]*wb[2];
  float z[3] = { dbw*wb[0], dbw*wb[1], dbw*wb[2] };
  float zb[3]; norm3(z, zb);

  bool ok = valid[p] != 0;
  float* Rp = ws + OFF_R + (size_t)p * 9;
  float* tp = ws + OFF_T + (size_t)p * 3;
#pragma unroll
  for (int k = 0; k < 3; ++k) {
    Rp[0*3 + k] = ok ? xb[k] : (k == 0 ? 1.f : 0.f);
    Rp[1*3 + k] = ok ? yb[k] : (k == 1 ? 1.f : 0.f);
    Rp[2*3 + k] = ok ? zb[k] : (k == 2 ? 1.f : 0.f);
    tp[k]       = ok ? ca[k] : 0.f;
  }
}

// ------------------------------------------------------------------
// Kernel 2: projections (512 -> 3 per head, 5 matrices) + rotation
// into global frame; store k-padded 4-vectors + squared norms.
// one thread = (position, head)
// ------------------------------------------------------------------
__global__ void ra_proj_kernel(const float* __restrict__ x,
                               const float* __restrict__ Wqr, const float* __restrict__ bqr,
                               const float* __restrict__ Wkr, const float* __restrict__ bkr,
                               const float* __restrict__ Wqd, const float* __restrict__ bqd,
                               const float* __restrict__ Wkd, const float* __restrict__ bkd,
                               const float* __restrict__ Wv,  const float* __restrict__ bv,
                               float* __restrict__ ws) {
  int tid = blockIdx.x * blockDim.x + threadIdx.x;
  if (tid >= kPos * kHeads) return;
  int p = tid / kHeads;
  int h = tid % kHeads;
  int col = h * 3;

  float acc[15];
#pragma unroll
  for (int d = 0; d < 3; ++d) {
    acc[d]      = bqr[col + d];
    acc[3 + d]  = bkr[col + d];
    acc[6 + d]  = bqd[col + d];
    acc[9 + d]  = bkd[col + d];
    acc[12 + d] = bv[col + d];
  }
  const float* xp = x + (size_t)p * kDim;
  __builtin_prefetch(xp, 0, 0);  // global_prefetch_b8
  for (int c = 0; c < kDim; ++c) {
    float xv = xp[c];
    int base = c * kFused + col;
#pragma unroll
    for (int d = 0; d < 3; ++d) {
      acc[d]      = fmaf(xv, Wqr[base + d], acc[d]);
      acc[3 + d]  = fmaf(xv, Wkr[base + d], acc[3 + d]);
      acc[6 + d]  = fmaf(xv, Wqd[base + d], acc[6 + d]);
      acc[9 + d]  = fmaf(xv, Wkd[base + d], acc[9 + d]);
      acc[12 + d] = fmaf(xv, Wv[base + d],  acc[12 + d]);
    }
  }

  const float* Rp = ws + OFF_R + (size_t)p * 9;
  const float* tp = ws + OFF_T + (size_t)p * 3;
  int b = p / kL;
  int l = p % kL;
  size_t idx = ((size_t)(b * kHeads + h) * kL + l);

  float out[5][3];
#pragma unroll
  for (int m = 0; m < 5; ++m) {
#pragma unroll
    for (int k = 0; k < 3; ++k) {
      out[m][k] = acc[m*3 + 0] * Rp[0*3 + k]
                + acc[m*3 + 1] * Rp[1*3 + k]
                + acc[m*3 + 2] * Rp[2*3 + k];
    }
  }
  // qd_g / kd_g get the translation
#pragma unroll
  for (int k = 0; k < 3; ++k) {
    out[2][k] += tp[k];
    out[3][k] += tp[k];
  }
  float qq = out[2][0]*out[2][0] + out[2][1]*out[2][1] + out[2][2]*out[2][2];
  float kk = out[3][0]*out[3][0] + out[3][1]*out[3][1] + out[3][2]*out[3][2];

  float* dst[5] = { ws + OFF_QR, ws + OFF_KR, ws + OFF_QD, ws + OFF_KD, ws + OFF_V };
#pragma unroll
  for (int m = 0; m < 5; ++m) {
    float* dp = dst[m] + idx * 4;
    dp[0] = out[m][0]; dp[1] = out[m][1]; dp[2] = out[m][2]; dp[3] = 0.f;
  }
  ws[OFF_QQ + idx] = qq;
  ws[OFF_KK + idx] = kk;
}

// ------------------------------------------------------------------
// Kernel 3: fused flash attention core. One wave (32 lanes) per
// (b, h, 16-row i-block). Two V_WMMA_F32_16X16X4_F32 per 16x16 tile.
// K/V tiles are staged global->LDS with double-buffered async copies
// (ASYNCcnt pipeline: issue next tile, s_wait_asynccnt 2, compute).
//
// A layout (16x4 f32): lane m in [0,16): rows, VGPR pair = K {0,1};
//                      lanes 16..31: K {2,3}. -> per-lane float2.
// B layout (4x16 f32): mirrored (K split by lane half, N on lanes).
// C/D layout: lane n -> col N = n&15; VGPR g -> row M = g + 8*(n>>4).
// ------------------------------------------------------------------
__global__ __launch_bounds__(32)
void ra_attn_kernel(const unsigned char* __restrict__ valid,
                    const float* __restrict__ w_r,
                    const float* __restrict__ w_d,
                    float* __restrict__ ws) {
  constexpr int nIB = kL / 16;  // 96
  // per-buffer layout (floats): kr rows [0,64), kd rows [64,128), vg rows [128,192)
  __shared__ float smem[2][192];

  int bh = blockIdx.x / nIB;
  int ib = blockIdx.x % nIB;
  int b  = bh / kHeads;
  int h  = bh % kHeads;
  int i0 = ib * 16;

  int lane = threadIdx.x;
  int n    = lane & 15;
  int half = lane >> 4;
  int k0   = half * 2;

  const float* qr = ws + OFF_QR + (size_t)bh * kL * 4;
  const float* qd = ws + OFF_QD + (size_t)bh * kL * 4;
  const float* kr = ws + OFF_KR + (size_t)bh * kL * 4;
  const float* kd = ws + OFF_KD + (size_t)bh * kL * 4;
  const float* vg = ws + OFF_V  + (size_t)bh * kL * 4;
  const float* qq = ws + OFF_QQ + (size_t)bh * kL;
  const float* Rm = ws + OFF_R;
  float* obuf = ws + OFF_O;

  float wrv = w_r[h], wdv = w_d[h];
  float wr = (wrv > 20.f) ? wrv : log1pf(__expf(wrv));   // softplus
  float wd = (wdv > 20.f) ? wdv : log1pf(__expf(wdv));
  const float inv_sqrt3 = 0.57735026918962576f;

  // A-matrix fragments (queries), resident for the whole j-loop
  v2f a_qr, a_qd;
  {
    const float* pq = qr + (size_t)(i0 + n) * 4 + k0;
    a_qr[0] = pq[0]; a_qr[1] = pq[1];
    const float* pd = qd + (size_t)(i0 + n) * 4 + k0;
    a_qd[0] = pd[0]; a_qd[1] = pd[1];
  }
  float qqrow[8];
#pragma unroll
  for (int g = 0; g < 8; ++g) qqrow[g] = qq[i0 + g + 8 * half];

  float mrow[8], lrow[8], op0[8], op1[8], op2[8];
#pragma unroll
  for (int g = 0; g < 8; ++g) {
    mrow[g] = -1e30f; lrow[g] = 0.f;
    op0[g] = 0.f; op1[g] = 0.f; op2[g] = 0.f;
  }

  // --- async tile stage: 2 x b128 per wave (all 32 lanes active) ---
  // op1: lanes 0-15 stage kr row n; lanes 16-31 stage kd row n.
  // op2: all lanes stage vg row n (halves mirror; duplicate write is benign).
  auto issue_tile = [&](int buf, int j0) {
    const float* g1 = (half ? kd : kr) + (size_t)(j0 + n) * 4;
    unsigned int  l1 = lds_offset(&smem[buf][(half ? 64 : 0) + n * 4]);
    async_copy_b128(l1, g1);
    const float* g2 = vg + (size_t)(j0 + n) * 4;
    unsigned int  l2 = lds_offset(&smem[buf][128 + n * 4]);
    async_copy_b128(l2, g2);
  };

  issue_tile(0, 0);  // prologue: tile 0 in flight (ASYNCcnt = 2)

  for (int jb = 0; jb < nIB; ++jb) {
    int cur = jb & 1;
    if (jb + 1 < nIB) {
      issue_tile(cur ^ 1, (jb + 1) * 16);  // ASYNCcnt = 4
      wait_async_le2();                    // current tile's 2 copies landed
    } else {
      wait_async_0();                      // drain last tile
    }

    const float* bufc = &smem[cur][0];
    const float* pk  = bufc + n * 4;        // kr row n
    const float* pkd = bufc + 64 + n * 4;   // kd row n
    const float* pv  = bufc + 128 + n * 4;  // vg row n

    v2f b_kr, b_kd;
    b_kr[0] = pk[k0];  b_kr[1] = pk[k0 + 1];
    float kd0 = pkd[0], kd1 = pkd[1], kd2 = pkd[2];
    b_kd[0] = pkd[k0]; b_kd[1] = pkd[k0 + 1];
    float kkj = kd0*kd0 + kd1*kd1 + kd2*kd2;   // |kd_g[j]|^2 from staged tile
    float vx = pv[0], vy = pv[1], vz = pv[2];
    bool keyok = valid[(size_t)b * kL + jb * 16 + n] != 0;

    v8f cz = {0.f, 0.f, 0.f, 0.f, 0.f, 0.f, 0.f, 0.f};
    // dir = Qr_g (16x4) @ Kr_g^T (4x16)
    v8f dir = __builtin_amdgcn_wmma_f32_16x16x4_f32(
        false, a_qr, false, b_kr, (short)0, cz, false, false);
    // crs = Qd_g @ Kd_g^T  (cross term of squared distance)
    v8f crs = __builtin_amdgcn_wmma_f32_16x16x4_f32(
        false, a_qd, false, b_kd, (short)0, cz, false, false);

    float s[8], tm[8];
#pragma unroll
    for (int g = 0; g < 8; ++g) {
      float d2   = qqrow[g] + kkj - 2.f * crs[g];
      float dist = sqrtf(fmaxf(d2, 0.f));
      float sc   = (wr * dir[g] - wd * dist) * inv_sqrt3;
      s[g]  = keyok ? sc : -1e30f;
      tm[g] = s[g];
    }
    // row max over the 16 columns (columns live on lanes within a half)
#pragma unroll
    for (int mk = 1; mk <= 8; mk <<= 1)
#pragma unroll
      for (int g = 0; g < 8; ++g)
        tm[g] = fmaxf(tm[g], __shfl_xor(tm[g], mk, 32));

    float p[8], psum[8];
#pragma unroll
    for (int g = 0; g < 8; ++g) {
      float mn    = fmaxf(mrow[g], tm[g]);
      float scale = __expf(mrow[g] - mn);
      p[g]    = __expf(s[g] - mn);
      psum[g] = p[g];
      mrow[g] = mn;
      lrow[g] *= scale;
      op0[g] *= scale; op1[g] *= scale; op2[g] *= scale;
    }
#pragma unroll
    for (int mk = 1; mk <= 8; mk <<= 1)
#pragma unroll
      for (int g = 0; g < 8; ++g)
        psum[g] += __shfl_xor(psum[g], mk, 32);
#pragma unroll
    for (int g = 0; g < 8; ++g) {
      lrow[g] += psum[g];
      op0[g] = fmaf(p[g], vx, op0[g]);
      op1[g] = fmaf(p[g], vy, op1[g]);
      op2[g] = fmaf(p[g], vz, op2[g]);
    }
  }

  // reduce lane-partial P@V over the 16 columns (once, at the end)
#pragma unroll
  for (int mk = 1; mk <= 8; mk <<= 1)
#pragma unroll
    for (int g = 0; g < 8; ++g) {
      op0[g] += __shfl_xor(op0[g], mk, 32);
      op1[g] += __shfl_xor(op1[g], mk, 32);
      op2[g] += __shfl_xor(op2[g], mk, 32);
    }

  // lane 0 of each half writes its 8 rows: normalize, rotate back, store
  if (n == 0) {
#pragma unroll
    for (int g = 0; g < 8; ++g) {
      int row = i0 + g + 8 * half;
      float inv_l = (lrow[g] > 0.f) ? 1.f / lrow[g] : 0.f;
      float og0 = op0[g] * inv_l, og1 = op1[g] * inv_l, og2 = op2[g] * inv_l;
      size_t pp = (size_t)b * kL + row;
      const float* Rp = Rm + pp * 9;
#pragma unroll
      for (int d = 0; d < 3; ++d) {
        float ol = og0 * Rp[0*3 + d] + og1 * Rp[1*3 + d] + og2 * Rp[2*3 + d];
        obuf[pp * kFused + h * 3 + d] = ol;
      }
    }
  }
}

// ------------------------------------------------------------------
// Kernel 4: output projection 24 -> 512, masked
// ------------------------------------------------------------------
__global__ void ra_outproj_kernel(const float* __restrict__ ws,
                                  const unsigned char* __restrict__ valid,
                                  const float* __restrict__ Wproj,
                                  const float* __restrict__ bproj,
                                  float* __restrict__ out) {
  int tid = blockIdx.x * blockDim.x + threadIdx.x;
  if (tid >= kPos * kDim) return;
  int p = tid / kDim;
  int d = tid % kDim;
  const float* o = ws + OFF_O + (size_t)p * kFused;
  float acc = bproj[d];
#pragma unroll
  for (int c = 0; c < kFused; ++c) acc = fmaf(o[c], Wproj[c * kDim + d], acc);
  out[tid] = valid[p] ? acc : 0.f;
}

// ------------------------------------------------------------------
extern "C" void kernel_launch(void* const* d_in, const int* in_sizes, int n_in,
                              void* d_out, int out_size, void* d_ws, size_t ws_size,
                              hipStream_t stream) {
  (void)in_sizes; (void)n_in; (void)out_size; (void)ws_size;
  const float* x       = (const float*)d_in[0];
  const float* coords  = (const float*)d_in[1];
  const unsigned char* content = (const unsigned char*)d_in[2];
  // d_in[3] (mask) unused by reference math
  const float* Wqr = (const float*)d_in[4];   const float* bqr = (const float*)d_in[5];
  const float* Wkr = (const float*)d_in[6];   const float* bkr = (const float*)d_in[7];
  const float* Wqd = (const float*)d_in[8];   const float* bqd = (const float*)d_in[9];
  const float* Wkd = (const float*)d_in[10];  const float* bkd = (const float*)d_in[11];
  const float* Wv  = (const float*)d_in[12];  const float* bv  = (const float*)d_in[13];
  const float* w_r = (const float*)d_in[14];  const float* w_d = (const float*)d_in[15];
  const float* Wproj = (const float*)d_in[16];
  const float* bproj = (const float*)d_in[17];
  float* out = (float*)d_out;
  float* ws  = (float*)d_ws;

  { // frames
    int threads = 128, blocks = (kPos + threads - 1) / threads;
    ra_frames_kernel<<<blocks, threads, 0, stream>>>(coords, content, ws);
  }
  { // projections + rotation
    int total = kPos * kHeads;
    int threads = 128, blocks = (total + threads - 1) / threads;
    ra_proj_kernel<<<blocks, threads, 0, stream>>>(
        x, Wqr, bqr, Wkr, bkr, Wqd, bqd, Wkd, bkd, Wv, bv, ws);
  }
  { // fused WMMA flash attention with async LDS staging
    int blocks = kBH * (kL / 16);  // 1536 waves
    ra_attn_kernel<<<blocks, 32, 0, stream>>>(content, w_r, w_d, ws);
  }
  { // output projection
    int total = kPos * kDim;
    int threads = 256, blocks = (total + threads - 1) / threads;
    ra_outproj_kernel<<<blocks, threads, 0, stream>>>(ws, content, Wproj, bproj, out);
  }
}